// GradientBasedNoiseMask_20770461843630
// MI455X (gfx1250) — compile-verified
//
#include <hip/hip_runtime.h>
#include <hip/hip_bf16.h>
#include <math.h>

// Problem constants (match reference)
#define BB 2048
#define DD 8192
#define CC 1000
#define NOISE_SCALE     0.1f
#define NOISE_RATIO     0.3f
#define ADAPTIVE_FACTOR 1.0f
#define MAX_NOISE       1.0f

typedef __attribute__((ext_vector_type(2))) float v2f;
typedef __attribute__((ext_vector_type(4))) float v4f;
typedef __attribute__((ext_vector_type(8))) float v8f;

// ---------------------------------------------------------------------------
// Kernel 1: per-row cross-entropy loss for 16 rows per wave, using
// V_WMMA_F32_16X16X4_F32 with B = ones to do the sum(exp) reduction on the
// matrix pipe.  C = 1000 = 250 * 4, so no K tail.
// A-layout (ISA 7.12.2, 32-bit A 16x4): lanes 0-15 hold (M=lane, K=0/1) in
// VGPR0/1; lanes 16-31 hold (M=lane-16, K=2/3).
// D-layout (32-bit C/D 16x16): VGPR j, lanes 0-15 -> M=j; lanes 16-31 -> M=j+8.
// B = all ones makes every column of D equal to the row sum, so the exact
// B layout is irrelevant.
// ---------------------------------------------------------------------------
__global__ __launch_bounds__(32)
void ce_loss_wmma(const float* __restrict__ logits,
                  const int*   __restrict__ labels,
                  float*       __restrict__ loss_out) {
  const int lane = threadIdx.x;           // 0..31, wave32
  const int r0   = blockIdx.x * 16;       // 16 rows per wave
  const int m    = lane & 15;             // my row within the tile
  const int row  = r0 + m;
  const float* rowp = logits + (long)row * CC;

  // ---- pass 1: row max; lanes<16 scan even k, lanes>=16 scan odd k ----
  float mx = -INFINITY;
  for (int k = (lane >> 4); k < CC; k += 2)
    mx = fmaxf(mx, rowp[k]);
  mx = fmaxf(mx, __shfl_xor(mx, 16, 32)); // both half-waves now hold row max

  // logit at the label (duplicated across half-waves; harmless)
  const int lbl = labels[row];
  const float gl = rowp[lbl];

  // ---- pass 2: rowsum = sum_k exp(v - mx) via WMMA accumulate ----
  const int koff = (lane >> 4) * 2;       // 0 for lanes 0-15, 2 for lanes 16-31
  v8f c = (v8f){0.f,0.f,0.f,0.f,0.f,0.f,0.f,0.f};
  const v2f bones = (v2f){1.0f, 1.0f};
  for (int t = 0; t < CC; t += 4) {
    v2f a;
    a.x = __expf(rowp[t + koff]     - mx);
    a.y = __expf(rowp[t + koff + 1] - mx);
    // 8 args: (neg_a, A, neg_b, B, c_mod, C, reuse_a, reuse_b)
    c = __builtin_amdgcn_wmma_f32_16x16x4_f32(false, a, false, bones,
                                              (short)0, c, false, false);
  }

  // Extract row sums with compile-time vector indices + cross-lane broadcast:
  // row j   lives in c[j] of lanes 0-15; row j+8 in c[j] of lanes 16-31.
  float rowsum = 0.0f;
#pragma unroll
  for (int j = 0; j < 8; ++j) {
    const float s_lo = __shfl(c[j], 0, 32);   // row j
    const float s_hi = __shfl(c[j], 16, 32);  // row j+8
    if (lane == j)     rowsum = s_lo;
    if (lane == j + 8) rowsum = s_hi;
  }

  if (lane < 16)
    loss_out[row] = __logf(rowsum) + mx - gl;   // -log_softmax at label
}

// ---------------------------------------------------------------------------
// Kernel 2: min/max over the 2048 per-sample losses (single workgroup).
// ---------------------------------------------------------------------------
__global__ __launch_bounds__(256)
void loss_minmax(const float* __restrict__ loss,
                 float*       __restrict__ mm) {
  __shared__ float smin[256];
  __shared__ float smax[256];
  const int t = threadIdx.x;
  float lo = INFINITY, hi = -INFINITY;
  for (int i = t; i < BB; i += 256) {
    const float v = loss[i];
    lo = fminf(lo, v);
    hi = fmaxf(hi, v);
  }
  smin[t] = lo; smax[t] = hi;
  __syncthreads();
  for (int s = 128; s > 0; s >>= 1) {
    if (t < s) {
      smin[t] = fminf(smin[t], smin[t + s]);
      smax[t] = fmaxf(smax[t], smax[t + s]);
    }
    __syncthreads();
  }
  if (t == 0) { mm[0] = smin[0]; mm[1] = smax[0]; }
}

// ---------------------------------------------------------------------------
// Kernel 3: fused masked-noise apply.  HBM-bound streaming: 192 MB in,
// 64 MB out.  b128 vector accesses with non-temporal hints so the one-shot
// stream doesn't thrash L2.  Each block covers 4096 contiguous floats of a
// single row (D = 8192 -> 2 blocks/row), so `scale` is block-uniform.
// ---------------------------------------------------------------------------
__global__ __launch_bounds__(256)
void apply_noise(const float* __restrict__ x,
                 const float* __restrict__ u,
                 const float* __restrict__ nz,
                 const float* __restrict__ loss,
                 const float* __restrict__ mm,
                 float*       __restrict__ out) {
  const int b = blockIdx.x >> 1;                    // row index
  const float lmin = mm[0];
  const float lmax = mm[1];
  const float lnorm = (loss[b] - lmin) / (lmax - lmin + 1e-8f);
  const float nw    = 1.0f - lnorm;
  const float scale = fminf(NOISE_SCALE * (1.0f + ADAPTIVE_FACTOR * nw),
                            MAX_NOISE);

  const v4f* xv = (const v4f*)x;
  const v4f* uv = (const v4f*)u;
  const v4f* nv = (const v4f*)nz;
  v4f*       ov = (v4f*)out;

  const long vbase = (long)blockIdx.x * 1024 + threadIdx.x;  // v4f index
#pragma unroll
  for (int j = 0; j < 4; ++j) {
    const long i = vbase + (long)j * 256;
    const v4f xd = __builtin_nontemporal_load(&xv[i]);
    const v4f ud = __builtin_nontemporal_load(&uv[i]);
    const v4f nd = __builtin_nontemporal_load(&nv[i]);
    v4f r;
#pragma unroll
    for (int c2 = 0; c2 < 4; ++c2)
      r[c2] = xd[c2] + ((ud[c2] < NOISE_RATIO) ? nd[c2] * scale : 0.0f);
    __builtin_nontemporal_store(r, &ov[i]);
  }
}

// ---------------------------------------------------------------------------
// Host-side launcher (graph-capture safe: no mallocs/syncs, stream only).
// Workspace layout: [0 .. BB) per-sample loss, [BB, BB+2) {min, max}.
// ---------------------------------------------------------------------------
extern "C" void kernel_launch(void* const* d_in, const int* in_sizes, int n_in,
                              void* d_out, int out_size, void* d_ws, size_t ws_size,
                              hipStream_t stream) {
  const float* x      = (const float*)d_in[0];   // [B, D]
  const float* mo     = (const float*)d_in[1];   // [B, C]
  const int*   labels = (const int*)  d_in[2];   // [B]
  const float* u      = (const float*)d_in[3];   // [B, D]
  const float* nz     = (const float*)d_in[4];   // [B, D]
  float* out = (float*)d_out;

  float* ws_loss = (float*)d_ws;      // BB floats
  float* mm      = ws_loss + BB;      // 2 floats

  ce_loss_wmma<<<BB / 16, 32, 0, stream>>>(mo, labels, ws_loss);
  loss_minmax<<<1, 256, 0, stream>>>(ws_loss, mm);
  apply_noise<<<(BB * DD) / 4096, 256, 0, stream>>>(x, u, nz, ws_loss, mm, out);
}